// VQAModel_Pretrained_Attention_75892072120726
// MI455X (gfx1250) — compile-verified
//
#include <hip/hip_runtime.h>
#include <math.h>

// Problem dims (fixed by the reference)
constexpr int Vv = 20000, Bb = 128, Nn = 49, Ff = 2048;
constexpr int LQ = 20, TA = 20, Ee = 300, Hh = 512, Cc = 256;

typedef float v2f __attribute__((ext_vector_type(2)));
typedef float v8f __attribute__((ext_vector_type(8)));

// ---------------------------------------------------------------------------
// Generic fp32 WMMA GEMM:  D = A[M x K] * W[N x K]^T  (+ bias | += D)
//   - A row may be indirected through rowidx (fused embedding gather)
//   - W may start at a column offset wcol inside a row of leading dim ldw
//   - grid = (N/32, M/32), block = 128 threads (4 waves, each one 16x16 tile)
// Requires: M % 32 == 0, N % 32 == 0, K % 4 == 0 (true for all uses here).
// ---------------------------------------------------------------------------
__launch_bounds__(128)
__global__ void gemm_wmma_f32(const float* __restrict__ A, int lda,
                              const int* __restrict__ rowidx, int ridx_stride, int ridx_off,
                              const float* __restrict__ W, int ldw, int wcol,
                              const float* __restrict__ bias,
                              float* __restrict__ D, int ldd,
                              int K, int accumulate)
{
    __shared__ float As[32][17];
    __shared__ float Bs[32][17];

    const int tid  = threadIdx.x;
    const int wave = tid >> 5;
    const int lane = tid & 31;
    const int m0   = blockIdx.y * 32;
    const int n0   = blockIdx.x * 32;
    const int moff = (wave >> 1) << 4;   // 0 or 16
    const int noff = (wave & 1) << 4;    // 0 or 16
    const int lm   = lane & 15;
    const int hi   = lane >> 4;          // 0 or 1

    // cooperative-load mapping: 128 threads -> 32 rows x 4 float4 slots
    const int lr = tid >> 2;
    const int lq = tid & 3;

    const float* aRow;
    {
        const int gr = m0 + lr;
        const long ar = rowidx ? (long)rowidx[(long)gr * ridx_stride + ridx_off] : (long)gr;
        aRow = A + ar * (long)lda;
    }
    const float* wRow = W + (long)(n0 + lr) * ldw + wcol;

    v8f c;
    if (accumulate) {
#pragma unroll
        for (int v = 0; v < 8; ++v)
            c[v] = D[(long)(m0 + moff + v + 8 * hi) * ldd + (n0 + noff + lm)];
    } else if (bias) {
        const float bv = bias[n0 + noff + lm];
#pragma unroll
        for (int v = 0; v < 8; ++v) c[v] = bv;
    } else {
#pragma unroll
        for (int v = 0; v < 8; ++v) c[v] = 0.0f;
    }

    for (int k0 = 0; k0 < K; k0 += 16) {
        float4 av = make_float4(0.f, 0.f, 0.f, 0.f);
        float4 wv = make_float4(0.f, 0.f, 0.f, 0.f);
        const int kc = k0 + 4 * lq;
        if (kc < K) {                       // K % 4 == 0 -> full float4 valid
            av = *(const float4*)(aRow + kc);
            wv = *(const float4*)(wRow + kc);
        }
        As[lr][4 * lq + 0] = av.x; As[lr][4 * lq + 1] = av.y;
        As[lr][4 * lq + 2] = av.z; As[lr][4 * lq + 3] = av.w;
        Bs[lr][4 * lq + 0] = wv.x; Bs[lr][4 * lq + 1] = wv.y;
        Bs[lr][4 * lq + 2] = wv.z; Bs[lr][4 * lq + 3] = wv.w;
        __syncthreads();

#pragma unroll
        for (int kk = 0; kk < 16; kk += 4) {
            // ISA 7.12.2 fp32 16x4 A layout: lane L -> m = L%16, k = v + 2*(L>=16)
            v2f a, b;
            a.x = As[moff + lm][kk + 2 * hi];
            a.y = As[moff + lm][kk + 2 * hi + 1];
            b.x = Bs[noff + lm][kk + 2 * hi];
            b.y = Bs[noff + lm][kk + 2 * hi + 1];
            c = __builtin_amdgcn_wmma_f32_16x16x4_f32(
                    false, a, false, b, (short)0, c, false, false);
        }
        __syncthreads();
    }

#pragma unroll
    for (int v = 0; v < 8; ++v)
        D[(long)(m0 + moff + v + 8 * hi) * ldd + (n0 + noff + lm)] = c[v];
}

// ---------------------------------------------------------------------------
// LSTM cell pointwise: gates (B x 4H, order i|f|g|o), biases added here.
// h,c updated in place. idx over B*H.
// ---------------------------------------------------------------------------
__global__ void lstm_cell_k(const float* __restrict__ gates,
                            const float* __restrict__ bih,
                            const float* __restrict__ bhh,
                            float* __restrict__ h, float* __restrict__ c)
{
    const int idx = blockIdx.x * blockDim.x + threadIdx.x;  // 0 .. B*H-1
    const int b = idx >> 9;          // / 512
    const int j = idx & (Hh - 1);    // % 512
    const float* g = gates + (long)b * (4 * Hh);

    const float gi = g[j]            + bih[j]            + bhh[j];
    const float gf = g[Hh + j]       + bih[Hh + j]       + bhh[Hh + j];
    const float gg = g[2 * Hh + j]   + bih[2 * Hh + j]   + bhh[2 * Hh + j];
    const float go = g[3 * Hh + j]   + bih[3 * Hh + j]   + bhh[3 * Hh + j];

    const float si = 1.0f / (1.0f + expf(-gi));
    const float sf = 1.0f / (1.0f + expf(-gf));
    const float so = 1.0f / (1.0f + expf(-go));

    const float cn = sf * c[idx] + si * tanhf(gg);
    c[idx] = cn;
    h[idx] = so * tanhf(cn);
}

// ---------------------------------------------------------------------------
// Attention: per batch element (block), compute
//   scores[n] = sum_j tanh(feat_proj[b,n,j] + hW[b,j]) * attn_v[j]
//   alpha     = softmax(scores) over n (Nn=49)
//   context[b,c] = sum_n alpha[n] * feat[b,n,c]
// block = 256 threads; hW already contains attn_b.
// ---------------------------------------------------------------------------
__global__ void attn_k(const float* __restrict__ feat,
                       const float* __restrict__ feat_proj,
                       const float* __restrict__ hW,
                       const float* __restrict__ attn_v,
                       float* __restrict__ context)
{
    __shared__ float red[256];
    __shared__ float sc[Nn];

    const int b   = blockIdx.x;
    const int tid = threadIdx.x;
    const float* hwb = hW + (long)b * Hh;

    for (int n = 0; n < Nn; ++n) {
        const float* fp = feat_proj + (long)(b * Nn + n) * Hh;
        float p = 0.f;
        for (int j = tid; j < Hh; j += 256)
            p += tanhf(fp[j] + hwb[j]) * attn_v[j];
        red[tid] = p;
        __syncthreads();
        for (int s = 128; s > 0; s >>= 1) {
            if (tid < s) red[tid] += red[tid + s];
            __syncthreads();
        }
        if (tid == 0) sc[n] = red[0];
        __syncthreads();
    }

    if (tid == 0) {
        float mx = sc[0];
        for (int n = 1; n < Nn; ++n) mx = fmaxf(mx, sc[n]);
        float s = 0.f;
        for (int n = 0; n < Nn; ++n) { const float e = expf(sc[n] - mx); sc[n] = e; s += e; }
        const float inv = 1.0f / s;
        for (int n = 0; n < Nn; ++n) sc[n] *= inv;
    }
    __syncthreads();

    // Cc == 256 == blockDim.x
    float acc = 0.f;
    const float* fb = feat + (long)b * Nn * Cc;
    for (int n = 0; n < Nn; ++n) acc += sc[n] * fb[n * Cc + tid];
    context[(long)b * Cc + tid] = acc;
}

__global__ void zero_f32(float* __restrict__ p, int n)
{
    const int i = blockIdx.x * blockDim.x + threadIdx.x;
    if (i < n) p[i] = 0.0f;
}

// ---------------------------------------------------------------------------
extern "C" void kernel_launch(void* const* d_in, const int* in_sizes, int n_in,
                              void* d_out, int out_size, void* d_ws, size_t ws_size,
                              hipStream_t stream)
{
    (void)in_sizes; (void)n_in; (void)out_size; (void)ws_size;

    const float* img      = (const float*)d_in[0];
    const int*   ques     = (const int*)  d_in[1];
    const int*   ans      = (const int*)  d_in[2];
    const float* reduce_W = (const float*)d_in[3];
    const float* reduce_b = (const float*)d_in[4];
    const float* q_emb    = (const float*)d_in[5];
    const float* q_Wih    = (const float*)d_in[6];
    const float* q_Whh    = (const float*)d_in[7];
    const float* q_bih    = (const float*)d_in[8];
    const float* q_bhh    = (const float*)d_in[9];
    const float* attn_W   = (const float*)d_in[10];
    const float* attn_b   = (const float*)d_in[11];
    const float* attn_v   = (const float*)d_in[12];
    const float* dec_emb  = (const float*)d_in[13];
    const float* dec_Wih  = (const float*)d_in[14];
    const float* dec_Whh  = (const float*)d_in[15];
    const float* dec_bih  = (const float*)d_in[16];
    const float* dec_bhh  = (const float*)d_in[17];
    const float* fc_W     = (const float*)d_in[18];
    const float* fc_b     = (const float*)d_in[19];
    float*       out      = (float*)d_out;

    // workspace carve-up (floats)
    float* ws        = (float*)d_ws;
    float* feat      = ws;                                  // B*N*C
    float* feat_proj = feat      + (long)Bb * Nn * Cc;      // B*N*H
    float* gates     = feat_proj + (long)Bb * Nn * Hh;      // B*4H
    float* h         = gates     + (long)Bb * 4 * Hh;       // B*H
    float* c         = h         + (long)Bb * Hh;           // B*H
    float* hW        = c         + (long)Bb * Hh;           // B*H
    float* context   = hW        + (long)Bb * Hh;           // B*C

    const dim3 blk(128);
    const dim3 gFeat(Cc / 32, (Bb * Nn) / 32);      // 8 x 196
    const dim3 gProj(Hh / 32, (Bb * Nn) / 32);      // 16 x 196
    const dim3 gGate(4 * Hh / 32, Bb / 32);         // 64 x 4
    const dim3 gHW(Hh / 32, Bb / 32);               // 16 x 4
    const dim3 gFC(Vv / 32, Bb / 32);               // 625 x 4

    // 1) feat = img @ reduce_W.T + reduce_b
    gemm_wmma_f32<<<gFeat, blk, 0, stream>>>(img, Ff, nullptr, 0, 0,
                                             reduce_W, Ff, 0, reduce_b,
                                             feat, Cc, Ff, 0);
    // 2) feat_proj = feat @ Wf.T   (Wf = attn_W[:, :C])
    gemm_wmma_f32<<<gProj, blk, 0, stream>>>(feat, Cc, nullptr, 0, 0,
                                             attn_W, Cc + Hh, 0, nullptr,
                                             feat_proj, Hh, Cc, 0);
    // 3) h = c = 0
    zero_f32<<<(2 * Bb * Hh + 255) / 256, 256, 0, stream>>>(h, 2 * Bb * Hh);

    // 4) question LSTM (20 steps)
    for (int t = 0; t < LQ; ++t) {
        // gates = q_emb[ques[:,t]] @ q_Wih.T   (gather fused via rowidx)
        gemm_wmma_f32<<<gGate, blk, 0, stream>>>(q_emb, Ee, ques, LQ, t,
                                                 q_Wih, Ee, 0, nullptr,
                                                 gates, 4 * Hh, Ee, 0);
        // gates += h @ q_Whh.T
        gemm_wmma_f32<<<gGate, blk, 0, stream>>>(h, Hh, nullptr, 0, 0,
                                                 q_Whh, Hh, 0, nullptr,
                                                 gates, 4 * Hh, Hh, 1);
        lstm_cell_k<<<(Bb * Hh) / 256, 256, 0, stream>>>(gates, q_bih, q_bhh, h, c);
    }

    // 5) decoder initial cell state = 0 (h carries q_h)
    zero_f32<<<(Bb * Hh + 255) / 256, 256, 0, stream>>>(c, Bb * Hh);

    // 6) decoder (20 steps)
    for (int t = 0; t < TA; ++t) {
        // hW = h @ Wh.T + attn_b   (Wh = attn_W[:, C:])
        gemm_wmma_f32<<<gHW, blk, 0, stream>>>(h, Hh, nullptr, 0, 0,
                                               attn_W, Cc + Hh, Cc, attn_b,
                                               hW, Hh, Hh, 0);
        // attention -> context
        attn_k<<<Bb, 256, 0, stream>>>(feat, feat_proj, hW, attn_v, context);
        // gates  = dec_emb[ans[:,t]] @ dec_Wih[:, :E].T   (gather fused)
        gemm_wmma_f32<<<gGate, blk, 0, stream>>>(dec_emb, Ee, ans, TA, t,
                                                 dec_Wih, Ee + Cc, 0, nullptr,
                                                 gates, 4 * Hh, Ee, 0);
        // gates += context @ dec_Wih[:, E:].T
        gemm_wmma_f32<<<gGate, blk, 0, stream>>>(context, Cc, nullptr, 0, 0,
                                                 dec_Wih, Ee + Cc, Ee, nullptr,
                                                 gates, 4 * Hh, Cc, 1);
        // gates += h @ dec_Whh.T
        gemm_wmma_f32<<<gGate, blk, 0, stream>>>(h, Hh, nullptr, 0, 0,
                                                 dec_Whh, Hh, 0, nullptr,
                                                 gates, 4 * Hh, Hh, 1);
        lstm_cell_k<<<(Bb * Hh) / 256, 256, 0, stream>>>(gates, dec_bih, dec_bhh, h, c);
        // logits[:, t, :] = h @ fc_W.T + fc_b   (stored straight into d_out)
        gemm_wmma_f32<<<gFC, blk, 0, stream>>>(h, Hh, nullptr, 0, 0,
                                               fc_W, Hh, 0, fc_b,
                                               out + (long)t * Vv, TA * Vv, Hh, 0);
    }
}